// GATBlock_14353780704102
// MI455X (gfx1250) — compile-verified
//
#include <hip/hip_runtime.h>
#include <hip/hip_bf16.h>
#include <math.h>

// ---------------------------------------------------------------------------
// GATv2 block for MI455X (gfx1250, wave32).
// GEMMs use V_WMMA_F32_16X16X4_F32 (full fp32 precision, matches reference).
// One wave computes a 32x64 output strip: 2 A fragments x 4 B fragments ->
// 8 WMMAs per K-quad, amortizing the 8 scalar B loads (different W rows,
// cannot be vector-widened) over twice the math.
// Edge phase: fused edge-attr transform + leaky-relu + attention dot,
// segment-softmax via ordered-uint atomicMax + fp32 atomicAdd (L2-resident).
// ---------------------------------------------------------------------------

typedef __attribute__((ext_vector_type(2))) float v2f;
typedef __attribute__((ext_vector_type(8))) float v8f;

__device__ __forceinline__ float wave_sum(float v) {
    for (int off = 16; off > 0; off >>= 1) v += __shfl_xor(v, off, 32);
    return v;
}

// Order-preserving float<->uint encoding for atomicMax on floats.
__device__ __forceinline__ unsigned enc_f32(float f) {
    unsigned u = __float_as_uint(f);
    return (u & 0x80000000u) ? ~u : (u | 0x80000000u);
}
__device__ __forceinline__ float dec_f32(unsigned u) {
    return (u & 0x80000000u) ? __uint_as_float(u & 0x7fffffffu)
                             : __uint_as_float(~u);
}

// ---------------------------------------------------------------------------
// C[M,Ncols] = A[M,K] @ W[K,Ncols], fp32, via v_wmma_f32_16x16x4_f32.
// One wave per 32x64 output strip. Requires M%32==0, K%4==0, Ncols%64==0.
// ISA 7.12.2 32-bit A 16x4 layout: lane<16 -> M=lane, K=k0,k0+1 in v[0],v[1];
// lane>=16 -> same M, K=k0+2,k0+3.  B mirrored; C/D: vgpr i holds row
// i + 8*(lane>>4), col lane&15.
// ---------------------------------------------------------------------------
__global__ __launch_bounds__(32) void gemm32x64_f32wmma(
    const float* __restrict__ A, const float* __restrict__ W,
    float* __restrict__ C, int M, int K, int Ncols)
{
    const int lane = threadIdx.x & 31;
    const int m0   = (blockIdx.x << 5) + (lane & 15);   // rows 0..15 of strip
    const int kk   = (lane >> 4) << 1;                  // 0 or 2
    const int nb   = blockIdx.y << 6;
    const int n    = nb + (lane & 15);

    v8f acc00 = {}; v8f acc01 = {}; v8f acc02 = {}; v8f acc03 = {};
    v8f acc10 = {}; v8f acc11 = {}; v8f acc12 = {}; v8f acc13 = {};
    const float* arow0 = A + (size_t)m0 * K;
    const float* arow1 = arow0 + (size_t)16 * K;        // rows 16..31

    for (int k0 = 0; k0 < K; k0 += 4) {
        v2f a0; a0.x = arow0[k0 + kk]; a0.y = arow0[k0 + kk + 1];
        v2f a1; a1.x = arow1[k0 + kk]; a1.y = arow1[k0 + kk + 1];
        const float* w0 = W + (size_t)(k0 + kk) * Ncols + n;
        const float* w1 = w0 + Ncols;
        v2f b0; b0.x = w0[0];  b0.y = w1[0];
        v2f b1; b1.x = w0[16]; b1.y = w1[16];
        v2f b2; b2.x = w0[32]; b2.y = w1[32];
        v2f b3; b3.x = w0[48]; b3.y = w1[48];
        acc00 = __builtin_amdgcn_wmma_f32_16x16x4_f32(false, a0, false, b0, (short)0, acc00, false, false);
        acc01 = __builtin_amdgcn_wmma_f32_16x16x4_f32(false, a0, false, b1, (short)0, acc01, false, false);
        acc02 = __builtin_amdgcn_wmma_f32_16x16x4_f32(false, a0, false, b2, (short)0, acc02, false, false);
        acc03 = __builtin_amdgcn_wmma_f32_16x16x4_f32(false, a0, false, b3, (short)0, acc03, false, false);
        acc10 = __builtin_amdgcn_wmma_f32_16x16x4_f32(false, a1, false, b0, (short)0, acc10, false, false);
        acc11 = __builtin_amdgcn_wmma_f32_16x16x4_f32(false, a1, false, b1, (short)0, acc11, false, false);
        acc12 = __builtin_amdgcn_wmma_f32_16x16x4_f32(false, a1, false, b2, (short)0, acc12, false, false);
        acc13 = __builtin_amdgcn_wmma_f32_16x16x4_f32(false, a1, false, b3, (short)0, acc13, false, false);
    }

    const int mr = (blockIdx.x << 5) + ((lane >> 4) << 3);
    const int nc = lane & 15;
    for (int i = 0; i < 8; ++i) {
        size_t row = (size_t)(mr + i) * Ncols + nb + nc;
        C[row]      = acc00[i];
        C[row + 16] = acc01[i];
        C[row + 32] = acc02[i];
        C[row + 48] = acc03[i];
        size_t row2 = row + (size_t)16 * Ncols;
        C[row2]      = acc10[i];
        C[row2 + 16] = acc11[i];
        C[row2 + 32] = acc12[i];
        C[row2 + 48] = acc13[i];
    }
}

// ---------------------------------------------------------------------------
// Per-edge attention logits. One wave per edge. Folds edge_attr@We (K=3)
// inline. Writes logits[E,H] and atomicMax into nmax[N,H] (encoded uint).
// ---------------------------------------------------------------------------
__global__ __launch_bounds__(256) void edge_logits_kernel(
    const float* __restrict__ xl, const float* __restrict__ xr,
    const float* __restrict__ att, const float* __restrict__ We,
    const float* __restrict__ eattr, const int* __restrict__ src,
    const int* __restrict__ dst, float* __restrict__ logits,
    unsigned* __restrict__ nmax, int E, int H, int C)
{
    int wid = blockIdx.x * (blockDim.x >> 5) + (threadIdx.x >> 5);
    if (wid >= E) return;                  // wave-uniform exit
    int lane = threadIdx.x & 31;
    int s = src[wid], d = dst[wid];
    int HC = H * C;
    float ea0 = eattr[(size_t)wid * 3 + 0];
    float ea1 = eattr[(size_t)wid * 3 + 1];
    float ea2 = eattr[(size_t)wid * 3 + 2];
    const float* xls = xl + (size_t)s * HC;
    const float* xrd = xr + (size_t)d * HC;

    for (int h = 0; h < H; ++h) {
        float p = 0.f;
        int base = h * C;
        for (int c = lane; c < C; c += 32) {
            int idx = base + c;
            float ev = ea0 * We[idx] + ea1 * We[HC + idx] + ea2 * We[2 * HC + idx];
            float mv = xls[idx] + xrd[idx] + ev;
            mv = mv > 0.f ? mv : 0.2f * mv;       // leaky_relu(0.2)
            p += att[idx] * mv;
        }
        p = wave_sum(p);
        if (lane == 0) {
            logits[(size_t)wid * H + h] = p;
            atomicMax(&nmax[(size_t)d * H + h], enc_f32(p));
        }
    }
}

// exp(logit - segmax) in place; atomicAdd into segment sum.
__global__ void edge_exp_kernel(
    const int* __restrict__ dst, const unsigned* __restrict__ nmax,
    float* __restrict__ logits, float* __restrict__ nsum, int E, int H)
{
    int tid = blockIdx.x * blockDim.x + threadIdx.x;
    if (tid >= E * H) return;
    int e = tid / H, h = tid - e * H;
    int d = dst[e];
    float ex = expf(logits[tid] - dec_f32(nmax[(size_t)d * H + h]));
    logits[tid] = ex;
    atomicAdd(&nsum[(size_t)d * H + h], ex);
}

// agg[dst] += alpha * xl[src]. One wave per edge.
__global__ __launch_bounds__(256) void edge_aggregate_kernel(
    const float* __restrict__ xl, const float* __restrict__ logits,
    const float* __restrict__ nsum, const int* __restrict__ src,
    const int* __restrict__ dst, float* __restrict__ agg,
    int E, int H, int C)
{
    int wid = blockIdx.x * (blockDim.x >> 5) + (threadIdx.x >> 5);
    if (wid >= E) return;
    int lane = threadIdx.x & 31;
    int s = src[wid], d = dst[wid];
    int HC = H * C;
    float al[4];
    for (int h = 0; h < H; ++h)
        al[h] = logits[(size_t)wid * H + h] / (nsum[(size_t)d * H + h] + 1e-16f);
    const float* xls = xl + (size_t)s * HC;
    float* ad = agg + (size_t)d * HC;
    for (int idx = lane; idx < HC; idx += 32)
        atomicAdd(&ad[idx], al[idx / C] * xls[idx]);
}

__global__ void bias_relu_kernel(float* __restrict__ x,
                                 const float* __restrict__ b,
                                 int total, int HC)
{
    int t = blockIdx.x * blockDim.x + threadIdx.x;
    if (t >= total) return;
    float v = x[t] + b[t % HC];
    x[t] = v > 0.f ? v : 0.f;
}

// mean over 2 heads + bias + LayerNorm(128). One wave per node.
__global__ __launch_bounds__(256) void finalize_kernel(
    const float* __restrict__ agg2, const float* __restrict__ b2,
    const float* __restrict__ gamma, const float* __restrict__ beta,
    float* __restrict__ out, int N)
{
    int wid = blockIdx.x * (blockDim.x >> 5) + (threadIdx.x >> 5);
    if (wid >= N) return;
    int lane = threadIdx.x & 31;
    const float* a = agg2 + (size_t)wid * 256;
    float v[4]; float s = 0.f;
    for (int i = 0; i < 4; ++i) {
        int c = lane + 32 * i;
        v[i] = 0.5f * (a[c] + a[128 + c]) + b2[c];
        s += v[i];
    }
    s = wave_sum(s);
    float mu = s * (1.f / 128.f);
    float vs = 0.f;
    for (int i = 0; i < 4; ++i) { float dlt = v[i] - mu; vs += dlt * dlt; }
    vs = wave_sum(vs);
    float rstd = rsqrtf(vs * (1.f / 128.f) + 1e-5f);
    for (int i = 0; i < 4; ++i) {
        int c = lane + 32 * i;
        out[(size_t)wid * 128 + c] = (v[i] - mu) * rstd * gamma[c] + beta[c];
    }
}

__global__ void zero_u32_kernel(unsigned* __restrict__ p, size_t n)
{
    size_t t = (size_t)blockIdx.x * blockDim.x + threadIdx.x;
    if (t < n) p[t] = 0u;     // 0 bits == 0.0f; also below any enc_f32(real)
}

// ---------------------------------------------------------------------------
extern "C" void kernel_launch(void* const* d_in, const int* in_sizes, int n_in,
                              void* d_out, int out_size, void* d_ws, size_t ws_size,
                              hipStream_t stream)
{
    const float* z     = (const float*)d_in[0];
    const int*   ei    = (const int*)  d_in[1];
    const float* eattr = (const float*)d_in[2];
    const float* Wl1   = (const float*)d_in[3];
    const float* Wr1   = (const float*)d_in[4];
    const float* We1   = (const float*)d_in[5];
    const float* att1  = (const float*)d_in[6];
    const float* b1    = (const float*)d_in[7];
    const float* Wl2   = (const float*)d_in[8];
    const float* Wr2   = (const float*)d_in[9];
    const float* We2   = (const float*)d_in[10];
    const float* att2  = (const float*)d_in[11];
    const float* b2    = (const float*)d_in[12];
    const float* gamma = (const float*)d_in[13];
    const float* beta  = (const float*)d_in[14];
    float* out = (float*)d_out;

    const int IN = 256, HID = 128, H1 = 4, H2 = 2;
    const int HC1 = H1 * HID;   // 512
    const int HC2 = H2 * HID;   // 256
    int N = in_sizes[0] / IN;   // 20000 (multiple of 32)
    int E = in_sizes[1] / 2;    // 320000
    const int* src = ei;
    const int* dst = ei + E;

    // Workspace layout (fp32 words), ~130 MB total:
    float* XL1  = (float*)d_ws;                 // N*HC1
    float* XR1  = XL1 + (size_t)N * HC1;        // N*HC1
    float* HB   = XR1 + (size_t)N * HC1;        // N*HC1 (agg1 -> relu -> layer2 input)
    float* LOGI = HB  + (size_t)N * HC1;        // E*H1 (reused as E*H2)
    unsigned* NMAX = (unsigned*)(LOGI + (size_t)E * H1);   // N*H1
    float* NSUM = (float*)(NMAX + (size_t)N * H1);         // N*H1
    // Layer-2 reuse: XL2/XR2 overwrite XL1/XR1; AGG2 in free half of XL1 region.
    float* XL2  = XL1;
    float* XR2  = XR1;
    float* AGG2 = XL1 + (size_t)N * HC2;

    const int TB = 256;
    const int wavesPerBlk = TB >> 5;
    int eBlocks = (E + wavesPerBlk - 1) / wavesPerBlk;
    int nBlocks = (N + wavesPerBlk - 1) / wavesPerBlk;
    auto zblk = [](size_t n) { return (int)((n + 255) / 256); };

    // ---------------- Layer 1 (concat=True) ----------------
    gemm32x64_f32wmma<<<dim3(N / 32, HC1 / 64), 32, 0, stream>>>(z, Wl1, XL1, N, IN, HC1);
    gemm32x64_f32wmma<<<dim3(N / 32, HC1 / 64), 32, 0, stream>>>(z, Wr1, XR1, N, IN, HC1);

    zero_u32_kernel<<<zblk((size_t)N * H1), TB, 0, stream>>>(NMAX, (size_t)N * H1);
    zero_u32_kernel<<<zblk((size_t)N * H1), TB, 0, stream>>>((unsigned*)NSUM, (size_t)N * H1);
    zero_u32_kernel<<<zblk((size_t)N * HC1), TB, 0, stream>>>((unsigned*)HB, (size_t)N * HC1);

    edge_logits_kernel<<<eBlocks, TB, 0, stream>>>(XL1, XR1, att1, We1, eattr,
                                                   src, dst, LOGI, NMAX, E, H1, HID);
    edge_exp_kernel<<<(E * H1 + TB - 1) / TB, TB, 0, stream>>>(dst, NMAX, LOGI, NSUM, E, H1);
    edge_aggregate_kernel<<<eBlocks, TB, 0, stream>>>(XL1, LOGI, NSUM, src, dst, HB, E, H1, HID);
    bias_relu_kernel<<<zblk((size_t)N * HC1), TB, 0, stream>>>(HB, b1, N * HC1, HC1);

    // ---------------- Layer 2 (concat=False) ----------------
    gemm32x64_f32wmma<<<dim3(N / 32, HC2 / 64), 32, 0, stream>>>(HB, Wl2, XL2, N, HC1, HC2);
    gemm32x64_f32wmma<<<dim3(N / 32, HC2 / 64), 32, 0, stream>>>(HB, Wr2, XR2, N, HC1, HC2);

    zero_u32_kernel<<<zblk((size_t)N * H2), TB, 0, stream>>>(NMAX, (size_t)N * H2);
    zero_u32_kernel<<<zblk((size_t)N * H2), TB, 0, stream>>>((unsigned*)NSUM, (size_t)N * H2);
    zero_u32_kernel<<<zblk((size_t)N * HC2), TB, 0, stream>>>((unsigned*)AGG2, (size_t)N * HC2);

    edge_logits_kernel<<<eBlocks, TB, 0, stream>>>(XL2, XR2, att2, We2, eattr,
                                                   src, dst, LOGI, NMAX, E, H2, HID);
    edge_exp_kernel<<<(E * H2 + TB - 1) / TB, TB, 0, stream>>>(dst, NMAX, LOGI, NSUM, E, H2);
    edge_aggregate_kernel<<<eBlocks, TB, 0, stream>>>(XL2, LOGI, NSUM, src, dst, AGG2, E, H2, HID);

    // mean heads + bias + LayerNorm -> d_out
    finalize_kernel<<<nBlocks, TB, 0, stream>>>(AGG2, b2, gamma, beta, out, N);
}